// HeteroGNN_76330158785211
// MI455X (gfx1250) — compile-verified
//
#include <hip/hip_runtime.h>
#include <hip/hip_bf16.h>
#include <stdint.h>

// ---------------------------------------------------------------------------
// HeteroGNN (PyG TransformerConv, heads=1) for MI455X / gfx1250.
//  - Dense projections use V_WMMA_F32_16X16X4_F32 (fp32 WMMA, exact precision).
//    Each wave owns 4 M-tiles (M=64) so one B (weight) fragment feeds 4
//    independent WMMAs per K-step: 4x less weight-fragment traffic and a
//    deeper independent-WMMA window to hide load latency.
//  - Edge embedding e = ea @ We recomputed on the fly (ED=2) to save HBM traffic.
//  - Segment softmax via encoded-uint atomicMax + float atomicAdd.
// ---------------------------------------------------------------------------

typedef __attribute__((ext_vector_type(2))) float v2f;
typedef __attribute__((ext_vector_type(8))) float v8f;

#define HN 128          // hidden channels
#define NEG_INF_ENC 0x007fffffu   // enc(-inf)

__device__ __forceinline__ unsigned enc_f32(float v) {
    unsigned u = __float_as_uint(v);
    return (u & 0x80000000u) ? ~u : (u | 0x80000000u);
}
__device__ __forceinline__ float dec_f32(unsigned u) {
    return (u & 0x80000000u) ? __uint_as_float(u & 0x7fffffffu)
                             : __uint_as_float(~u);
}

// ------------------------- init / zero ------------------------------------
__global__ void init_u32(unsigned* __restrict__ p, unsigned val, size_t n) {
    size_t i = (size_t)blockIdx.x * blockDim.x + threadIdx.x;
    size_t stride = (size_t)gridDim.x * blockDim.x;
    for (; i < n; i += stride) p[i] = val;
}

// ------------------------- fp32 WMMA GEMM ---------------------------------
// Y[rows,128] = X[rows,din] @ W[din,128] + bias   (accumulate: Y += ...)
// One wave -> four 16x16 tiles stacked in M (M=64). blockDim=256 (8 waves)
// covers all 128 columns. rows are multiples of 64 (all node counts are).
//
// Fragment layouts (CDNA5 ISA 7.12.2, 32-bit):
//  A 16x4:  lane(m = lane&15, h = lane>>4) holds X[m][2h], X[m][2h+1]
//  B 4x16:  lane(n = lane&15, h)           holds W[2h][n], W[2h+1][n]
//  C 16x16: VGPR i, lane -> M = 8*h + i, N = lane&15
__global__ void gemm_wmma_f32(const float* __restrict__ X,
                              const float* __restrict__ W,
                              const float* __restrict__ bias,
                              float* __restrict__ Y,
                              int din, int accumulate) {
    const int lane = threadIdx.x & 31;
    const int wave = threadIdx.x >> 5;
    const int row0 = blockIdx.x << 6;     // 64 rows per block
    const int col0 = wave << 4;           // 16 cols per wave
    const int h    = lane >> 4;
    const int mn   = lane & 15;

    float bval = bias[col0 + mn];
    v8f acc[4];
#pragma unroll
    for (int mt = 0; mt < 4; ++mt)
#pragma unroll
        for (int i = 0; i < 8; ++i) acc[mt][i] = bval;

    const float* xbase = X + (size_t)(row0 + mn) * din + 2 * h;
    const float* wcol  = W + col0 + mn + (size_t)(2 * h) * HN;
    const size_t xtile = (size_t)16 * din;   // 16 rows between M-tiles

    for (int k = 0; k < din; k += 4) {
        v2f b;
        b.x = wcol[(size_t)k * HN];
        b.y = wcol[(size_t)k * HN + HN];
#pragma unroll
        for (int mt = 0; mt < 4; ++mt) {
            v2f a = *(const v2f*)(xbase + (size_t)mt * xtile + k);
            acc[mt] = __builtin_amdgcn_wmma_f32_16x16x4_f32(
                /*neg_a=*/false, a, /*neg_b=*/false, b,
                /*c_mod=*/(short)0, acc[mt],
                /*reuse_a=*/false, /*reuse_b=*/false);
        }
    }

#pragma unroll
    for (int mt = 0; mt < 4; ++mt) {
        float* yp = Y + (size_t)(row0 + 16 * mt + 8 * h) * HN + col0 + mn;
#pragma unroll
        for (int i = 0; i < 8; ++i) {
            size_t off = (size_t)i * HN;
            if (accumulate) yp[off] += acc[mt][i];
            else            yp[off]  = acc[mt][i];
        }
    }
}

// ------------------- edge pass 1: alpha + segment max ----------------------
// One wave per edge; each lane covers 4 of the 128 channels.
__global__ void edge_alpha_max(const int* __restrict__ src,
                               const int* __restrict__ dst,
                               const float* __restrict__ ea,   // [E,2]
                               const float* __restrict__ We,   // [2,128]
                               const float* __restrict__ q,
                               const float* __restrict__ k,
                               float* __restrict__ alpha,
                               unsigned* __restrict__ maxbuf, int nE) {
    int wid  = (int)(((size_t)blockIdx.x * blockDim.x + threadIdx.x) >> 5);
    int lane = threadIdx.x & 31;
    if (wid >= nE) return;
    int s = src[wid], d = dst[wid];
    float ea0 = ea[2 * wid], ea1 = ea[2 * wid + 1];
    const float4 qv = *(const float4*)(q + (size_t)d * HN + lane * 4);
    const float4 kv = *(const float4*)(k + (size_t)s * HN + lane * 4);
    const float4 w0 = *(const float4*)(We + lane * 4);
    const float4 w1 = *(const float4*)(We + HN + lane * 4);
    float sum = qv.x * (kv.x + ea0 * w0.x + ea1 * w1.x)
              + qv.y * (kv.y + ea0 * w0.y + ea1 * w1.y)
              + qv.z * (kv.z + ea0 * w0.z + ea1 * w1.z)
              + qv.w * (kv.w + ea0 * w0.w + ea1 * w1.w);
#pragma unroll
    for (int off = 16; off >= 1; off >>= 1) sum += __shfl_xor(sum, off, 32);
    if (lane == 0) {
        float a = sum * 0.08838834764831845f;  // 1/sqrt(128)
        alpha[wid] = a;
        atomicMax(maxbuf + d, enc_f32(a));
    }
}

// ------------------- edge pass 2: exp + segment sum ------------------------
__global__ void edge_exp_sum(const int* __restrict__ dst,
                             const unsigned* __restrict__ maxbuf,
                             float* __restrict__ alpha,
                             float* __restrict__ sumbuf, int nE) {
    int j = (int)((size_t)blockIdx.x * blockDim.x + threadIdx.x);
    if (j >= nE) return;
    int d = dst[j];
    float a = expf(alpha[j] - dec_f32(maxbuf[d]));
    alpha[j] = a;
    atomicAdd(sumbuf + d, a);
}

// ------------------- edge pass 3: normalize + scatter ----------------------
__global__ void edge_scatter(const int* __restrict__ src,
                             const int* __restrict__ dst,
                             const float* __restrict__ ea,
                             const float* __restrict__ We,
                             const float* __restrict__ v,
                             const float* __restrict__ alpha,
                             const float* __restrict__ sumbuf,
                             float* __restrict__ acc, int nE) {
    int wid  = (int)(((size_t)blockIdx.x * blockDim.x + threadIdx.x) >> 5);
    int lane = threadIdx.x & 31;
    if (wid >= nE) return;
    int s = src[wid], d = dst[wid];
    float w = alpha[wid] / (sumbuf[d] + 1e-16f);
    float ea0 = ea[2 * wid], ea1 = ea[2 * wid + 1];
    const float4 vv = *(const float4*)(v + (size_t)s * HN + lane * 4);
    const float4 w0 = *(const float4*)(We + lane * 4);
    const float4 w1 = *(const float4*)(We + HN + lane * 4);
    float* ap = acc + (size_t)d * HN + lane * 4;
    atomicAdd(ap + 0, (vv.x + ea0 * w0.x + ea1 * w1.x) * w);
    atomicAdd(ap + 1, (vv.y + ea0 * w0.y + ea1 * w1.y) * w);
    atomicAdd(ap + 2, (vv.z + ea0 * w0.z + ea1 * w1.z) * w);
    atomicAdd(ap + 3, (vv.w + ea0 * w0.w + ea1 * w1.w) * w);
}

// ---------------------------------------------------------------------------
extern "C" void kernel_launch(void* const* d_in, const int* in_sizes, int n_in,
                              void* d_out, int out_size, void* d_ws, size_t ws_size,
                              hipStream_t stream) {
    (void)n_in; (void)out_size; (void)ws_size;
    const int NCNT[4]     = {4096, 16384, 65536, 32768};     // SB,PV,PQ,NB
    const size_t NOFF[4]  = {0, 4096, 4096 + 16384, 4096 + 16384 + 65536};
    const int NTOT = 4096 + 16384 + 65536 + 32768;           // 118784
    const int NMAX = 65536;
    const int ES[14] = {0,0,0,1,1,2,1,2,3,2,3,1,2,3};        // src type per edge type
    const int ED_[14]= {1,2,3,2,3,3,0,0,0,1,2,1,2,3};        // dst type per edge type
    const int E = in_sizes[4] / 28;                          // [14,2,E]

    const int*   edge_idx  = (const int*)d_in[4];
    const float* edge_attr = (const float*)d_in[5];

    // Workspace layout
    float* x1     = (float*)d_ws;                  // layer-0 output: [NTOT,128]
    float* qbuf   = x1   + (size_t)NTOT * HN;      // [65536,128]
    float* kbuf   = qbuf + (size_t)NMAX * HN;
    float* vbuf   = kbuf + (size_t)NMAX * HN;
    float* alpha  = vbuf + (size_t)NMAX * HN;      // [E]
    unsigned* maxbuf = (unsigned*)(alpha + E);     // [65536]
    float* sumbuf = (float*)(maxbuf + NMAX);       // [65536]

    // Param leaves: insertion order per edge-type dict:
    //   0:Wq 1:bq 2:Wk 3:bk 4:Wv 5:bv 6:We 7:Ws 8:bs   (layer-major)
    auto leaf = [&](int l, int i, int j) -> const float* {
        return (const float*)d_in[6 + (l * 14 + i) * 9 + j];
    };

    for (int l = 0; l < 2; ++l) {
        const int din = (l == 0) ? 16 : HN;
        float* accOut = (l == 0) ? x1 : (float*)d_out;
        init_u32<<<2048, 256, 0, stream>>>((unsigned*)accOut, 0u, (size_t)NTOT * HN);

        for (int i = 0; i < 14; ++i) {
            const int s = ES[i], d = ED_[i];
            const float* xs = (l == 0) ? (const float*)d_in[s] : x1 + NOFF[s] * HN;
            const float* xd = (l == 0) ? (const float*)d_in[d] : x1 + NOFF[d] * HN;
            const float *Wq = leaf(l,i,0), *bq = leaf(l,i,1),
                        *Wk = leaf(l,i,2), *bk = leaf(l,i,3),
                        *Wv = leaf(l,i,4), *bv = leaf(l,i,5),
                        *We = leaf(l,i,6),
                        *Ws = leaf(l,i,7), *bs = leaf(l,i,8);
            const int* esrc = edge_idx + ((size_t)i * 2 + 0) * E;
            const int* edst = edge_idx + ((size_t)i * 2 + 1) * E;
            const float* ea = edge_attr + (size_t)i * E * 2;
            float* accd = accOut + NOFF[d] * HN;

            // Projections (fp32 WMMA, 64 rows per block).
            // Skip connection accumulates into accd.
            gemm_wmma_f32<<<NCNT[d] / 64, 256, 0, stream>>>(xd, Wq, bq, qbuf, din, 0);
            gemm_wmma_f32<<<NCNT[s] / 64, 256, 0, stream>>>(xs, Wk, bk, kbuf, din, 0);
            gemm_wmma_f32<<<NCNT[s] / 64, 256, 0, stream>>>(xs, Wv, bv, vbuf, din, 0);
            gemm_wmma_f32<<<NCNT[d] / 64, 256, 0, stream>>>(xd, Ws, bs, accd, din, 1);

            // Segment softmax scratch
            init_u32<<<(NCNT[d] + 255) / 256, 256, 0, stream>>>(maxbuf, NEG_INF_ENC, (size_t)NCNT[d]);
            init_u32<<<(NCNT[d] + 255) / 256, 256, 0, stream>>>((unsigned*)sumbuf, 0u, (size_t)NCNT[d]);

            const int wgrid = (E * 32 + 255) / 256;  // one wave per edge
            edge_alpha_max<<<wgrid, 256, 0, stream>>>(esrc, edst, ea, We, qbuf, kbuf,
                                                      alpha, maxbuf, E);
            edge_exp_sum<<<(E + 255) / 256, 256, 0, stream>>>(edst, maxbuf, alpha, sumbuf, E);
            edge_scatter<<<wgrid, 256, 0, stream>>>(esrc, edst, ea, We, vbuf,
                                                    alpha, sumbuf, accd, E);
        }
    }
}